// WindowedSelfAttention_9801115370179
// MI455X (gfx1250) — compile-verified
//
#include <hip/hip_runtime.h>
#include <hip/hip_bf16.h>

typedef __attribute__((ext_vector_type(2))) float v2f;
typedef __attribute__((ext_vector_type(8))) float v8f;

constexpr int BB = 2, TT = 2048, DD = 1024, HH = 16, WW = 65, HD = 64, RR = 32;
constexpr int QKV_N = 3 * DD;     // 3072
constexpr int MROWS = BB * TT;    // 4096
constexpr int QTILE = 64;         // queries per attention block
constexpr int ROWS  = QTILE + 2 * RR;  // 128 K/V rows per tile
constexpr int RSTR  = 68;         // LDS row stride (floats): 272B = 16B-aligned rows,
                                  // bank = (4*row + d) % 64 -> conflict-spread

// ---------------------------------------------------------------------------
// C[M,N] = A[M,K] @ W[N,K]^T  (Linear: y = x W^T), all f32 row-major.
// One wave -> 16x64 C strip (4 N-tiles share one A fragment per k-step).
// V_WMMA_F32_16X16X4_F32 fragment layouts:
//   A (16x4): lane m=lane&15; VGPR0/1 = A[m, kk+2*half +0/+1]
//   B (4x16): lane n=lane&15; VGPR0/1 = W[n, kk+2*half +0/+1]
//   C/D:      VGPR r -> M = r + 8*half, N = lane&15.
// ---------------------------------------------------------------------------
__global__ __launch_bounds__(256)
void gemm_xwT_f32_wmma(const float* __restrict__ A, const float* __restrict__ Wt,
                       float* __restrict__ C, int M, int N, int K) {
    const int lane = threadIdx.x & 31;
    const int wave = threadIdx.x >> 5;
    const int tilesN4 = N >> 6;                    // 64-wide N groups
    const int wid = blockIdx.x * 8 + wave;
    const int tM = wid / tilesN4;
    const int tN = wid % tilesN4;
    if (tM * 16 >= M) return;

    const int half = lane >> 4;
    const int l16  = lane & 15;
    const float* arow = A  + (size_t)(tM * 16 + l16) * K + 2 * half;
    const float* brow = Wt + (size_t)(tN * 64 + l16) * K + 2 * half;

    v8f acc[4];
#pragma unroll
    for (int j = 0; j < 4; ++j) acc[j] = (v8f){0.f,0.f,0.f,0.f,0.f,0.f,0.f,0.f};

#pragma unroll 2
    for (int kk = 0; kk < K; kk += 4) {
        v2f a  = *(const v2f*)(arow + kk);
        v2f b0 = *(const v2f*)(brow + kk);
        v2f b1 = *(const v2f*)(brow + (size_t)16 * K + kk);
        v2f b2 = *(const v2f*)(brow + (size_t)32 * K + kk);
        v2f b3 = *(const v2f*)(brow + (size_t)48 * K + kk);
        acc[0] = __builtin_amdgcn_wmma_f32_16x16x4_f32(false, a, false, b0, (short)0, acc[0], false, false);
        acc[1] = __builtin_amdgcn_wmma_f32_16x16x4_f32(false, a, false, b1, (short)0, acc[1], false, false);
        acc[2] = __builtin_amdgcn_wmma_f32_16x16x4_f32(false, a, false, b2, (short)0, acc[2], false, false);
        acc[3] = __builtin_amdgcn_wmma_f32_16x16x4_f32(false, a, false, b3, (short)0, acc[3], false, false);
    }

#pragma unroll
    for (int j = 0; j < 4; ++j) {
        float* crow = C + (size_t)(tM * 16 + 8 * half) * N + tN * 64 + j * 16 + l16;
#pragma unroll
        for (int r = 0; r < 8; ++r) crow[(size_t)r * N] = acc[j][r];
    }
}

// ---------------------------------------------------------------------------
// Banded attention: block = 64 queries of one (b,h).
// K/V window staged in LDS via GLOBAL_LOAD_ASYNC_TO_LDS_B128 (ASYNCcnt).
// qkv layout: [b*T+t][3072] with Q at col 0, K at 1024, V at 2048.
// Kernel has no static __shared__, so dynamic LDS starts at offset 0 and the
// hand-computed LDS byte offsets below match the compiler's smem addressing.
// ---------------------------------------------------------------------------
__global__ __launch_bounds__(QTILE)
void attn_window(const float* __restrict__ qkv, const unsigned char* __restrict__ mask,
                 float* __restrict__ attn) {
    extern __shared__ float smem[];
    float* Kl = smem;                                   // ROWS x RSTR @ byte 0
    float* Vl = smem + ROWS * RSTR;                     // ROWS x RSTR
    unsigned char* vflag = (unsigned char*)(smem + 2 * ROWS * RSTR);  // ROWS

    const int tid   = threadIdx.x;
    const int tiles = TT / QTILE;
    const int blk   = blockIdx.x;
    const int tile  = blk % tiles;
    const int h     = (blk / tiles) % HH;
    const int b     = blk / (tiles * HH);
    const int t0    = tile * QTILE;
    const size_t basebt = (size_t)b * TT;

    // validity flags (edge-replicate clamp; mask sampled at clamped index)
    for (int row = tid; row < ROWS; row += QTILE) {
        int p = t0 - RR + row;
        int src = p < 0 ? 0 : (p > TT - 1 ? TT - 1 : p);
        vflag[row] = (p >= 0 && p < TT && mask[b * TT + src] != 0) ? 1 : 0;
    }

    // async K/V window fill: global -> LDS, no VGPR bounce
    const float* kbase = qkv + basebt * QKV_N + DD + h * HD;      // K head base
    const float* vbase = kbase + DD;                              // V head base
    for (int idx = tid; idx < ROWS * (HD / 4); idx += QTILE) {
        int row = idx >> 4;                 // 16 float4 chunks per 64-float row
        int c4  = (idx & 15) * 4;
        int p = t0 - RR + row;
        int src = p < 0 ? 0 : (p > TT - 1 ? TT - 1 : p);
        unsigned int klds = (unsigned int)((row * RSTR + c4) * 4);
        unsigned int vlds = klds + (unsigned int)(ROWS * RSTR * 4);
        unsigned int goff = (unsigned int)(((size_t)src * QKV_N + c4) * 4);
        asm volatile("global_load_async_to_lds_b128 %0, %1, %2"
                     :: "v"(klds), "v"(goff), "s"(kbase) : "memory");
        asm volatile("global_load_async_to_lds_b128 %0, %1, %2"
                     :: "v"(vlds), "v"(goff), "s"(vbase) : "memory");
    }
    asm volatile("s_wait_asynccnt 0x0" ::: "memory");
    __syncthreads();

    const int t = t0 + tid;
    float q[HD];
    const float* qr = qkv + (basebt + t) * QKV_N + h * HD;
#pragma unroll
    for (int i = 0; i < HD / 4; ++i) {
        float4 qv = *(const float4*)(qr + 4 * i);
        q[4*i] = qv.x; q[4*i+1] = qv.y; q[4*i+2] = qv.z; q[4*i+3] = qv.w;
    }

    const float scale = 0.125f;  // 64^-0.5
    // pass 1: running max of masked scores
    float m = -3.0e38f;
    for (int w = 0; w < WW; ++w) {
        const float* kr = &Kl[(tid + w) * RSTR];
        float s = 0.f;
#pragma unroll
        for (int d = 0; d < HD; ++d) s = fmaf(q[d], kr[d], s);
        s = vflag[tid + w] ? s * scale : -1.0e9f;
        m = fmaxf(m, s);
    }
    // pass 2: unnormalized softmax-weighted V accumulation
    float acc[HD];
#pragma unroll
    for (int d = 0; d < HD; ++d) acc[d] = 0.f;
    float l = 0.f;
    for (int w = 0; w < WW; ++w) {
        const int row = tid + w;
        const float* kr = &Kl[row * RSTR];
        float s = 0.f;
#pragma unroll
        for (int d = 0; d < HD; ++d) s = fmaf(q[d], kr[d], s);
        s = vflag[row] ? s * scale : -1.0e9f;
        float e = __expf(s - m);
        l += e;
        const float* vr = &Vl[row * RSTR];
#pragma unroll
        for (int d = 0; d < HD; ++d) acc[d] = fmaf(e, vr[d], acc[d]);
    }
    const float inv = 1.f / l;
    float* orow = attn + (basebt + t) * DD + h * HD;
#pragma unroll
    for (int d = 0; d < HD; ++d) orow[d] = acc[d] * inv;
}

// ---------------------------------------------------------------------------
extern "C" void kernel_launch(void* const* d_in, const int* in_sizes, int n_in,
                              void* d_out, int out_size, void* d_ws, size_t ws_size,
                              hipStream_t stream) {
    (void)in_sizes; (void)n_in; (void)out_size; (void)ws_size;
    const float*         x      = (const float*)d_in[0];
    const unsigned char* mask   = (const unsigned char*)d_in[1];  // jax bool -> 1 byte
    const float*         w_qkv  = (const float*)d_in[2];
    const float*         w_out  = (const float*)d_in[3];
    float*               out    = (float*)d_out;

    float* qkv   = (float*)d_ws;                         // 4096*3072 f32 = 48 MB
    float* attnb = qkv + (size_t)MROWS * QKV_N;          // 4096*1024 f32 = 16 MB

    // 1) qkv = x @ w_qkv^T   (M=4096, N=3072, K=1024)
    {
        const int M = MROWS, N = QKV_N, K = DD;
        const int waves = (M / 16) * (N / 64);           // 12288
        gemm_xwT_f32_wmma<<<waves / 8, 256, 0, stream>>>(x, w_qkv, qkv, M, N, K);
    }
    // 2) banded attention -> attnb (B,T,D)
    {
        const int blocks = BB * HH * (TT / QTILE);       // 1024
        const size_t shbytes = (size_t)2 * ROWS * RSTR * sizeof(float) + ROWS;
        attn_window<<<blocks, QTILE, shbytes, stream>>>(qkv, mask, attnb);
    }
    // 3) out = attnb @ w_out^T   (M=4096, N=1024, K=1024)
    {
        const int M = MROWS, N = DD, K = DD;
        const int waves = (M / 16) * (N / 64);           // 4096
        gemm_xwT_f32_wmma<<<waves / 8, 256, 0, stream>>>(attnb, w_out, out, M, N, K);
    }
}